// SmoothDix_78211354460885
// MI455X (gfx1250) — compile-verified
//
#include <hip/hip_runtime.h>
#include <math.h>

// ---- problem constants (match reference) ----
#define NTT 2000
#define NXX 512
#define NBB 32
#define NZZ 600
#define DT_F   0.001f
#define DZ_F   10.0f
#define LAM_F  10.0f
#define VMIN_F 1200.0f
#define VMAX_F 6000.0f
#define EPS_F  1e-6f

constexpr int TPB = 256;            // one trace (b,x) per thread
constexpr int CH  = 20;             // t-rows per TDM chunk
constexpr int NCHUNK = NTT / CH;    // 100

typedef __attribute__((ext_vector_type(4))) unsigned int v4u;
typedef __attribute__((ext_vector_type(8))) int          v8i;
typedef __attribute__((ext_vector_type(4))) int          v4i;

#if defined(__gfx1250__) && __has_builtin(__builtin_amdgcn_tensor_load_to_lds)
#define USE_TDM 1
#else
#define USE_TDM 0
#endif

#if USE_TDM
// Issue a TDM load of a (tile_dim0=256 x, tile_dim1=CH t) float tile into LDS.
// D# per CDNA5 ISA: group0 = flags/lds/global addr,
// group1 = dims/strides, groups 2-3 zero (2D tensor). This toolchain
// (clang-23 / therock-10.0) uses the 6-arg builtin: the extra int32x8 group
// is zero-filled, cpol last.
__device__ __forceinline__ void tdm_issue(const float* gsrc, unsigned lds_off) {
  unsigned long long ga = (unsigned long long)(uintptr_t)gsrc;
  v4u g0;
  g0.x = 1u;                                        // count=1, user descriptor
  g0.y = lds_off;                                   // lds_addr (bytes)
  g0.z = (unsigned)(ga & 0xffffffffull);            // global_addr[31:0]
  g0.w = (unsigned)((ga >> 32) & 0x1ffffffu)        // global_addr[56:32]
       | (2u << 30);                                // type=2 ("image")
  v8i g1;
  g1[0] = (2 << 16);                                // data_size=4B; mask=0
  g1[1] = (TPB & 0xffff) << 16;                     // tensor_dim0 lo16
  g1[2] = ((TPB >> 16) & 0xffff) | ((CH & 0xffff) << 16); // dim0 hi | dim1 lo
  g1[3] = ((CH >> 16) & 0xffff) | ((TPB & 0xffff) << 16); // dim1 hi | tile_dim0
  g1[4] = (CH & 0xffff);                            // tile_dim1=CH, tile_dim2=0
  g1[5] = NXX;                                      // tensor_dim0_stride = 512
  g1[6] = 0;
  g1[7] = 0;
  v4i g2 = {0, 0, 0, 0};
  v4i g3 = {0, 0, 0, 0};
  v8i g4 = {0, 0, 0, 0, 0, 0, 0, 0};
  __builtin_amdgcn_tensor_load_to_lds(g0, g1, g2, g3, g4, 0);
}
#endif

__global__ __launch_bounds__(TPB)
void SmoothDix_78211354460885_kernel(const float* __restrict__ rms,
                                     float* __restrict__ dv,
                                     float* __restrict__ tv) {
  __shared__ float s_b[NTT];              // eliminated diagonal b'
  __shared__ float s_w[NTT];              // elimination weights w
  __shared__ float s_in[2][CH][TPB];      // TDM double buffer

  const int tid = threadIdx.x;
  const int blk = blockIdx.x;             // 64 blocks
  const int b   = blk >> 1;
  const int x0  = (blk & 1) * TPB;

  // ---- Thomas coefficients, closed form (constant-coefficient Mobius) ----
  // b'_i = mu1*(A + B r^i)/(A + B r^(i-1)),  r = mu2/mu1 < 1
  {
    const double S   = 1.0 + 2.0 * (double)LAM_F;          // 21
    const double P   = (double)LAM_F * (double)LAM_F;      // 100
    const double dsc = sqrt(S * S - 4.0 * P);              // sqrt(41)
    const double mu1 = 0.5 * (S + dsc);
    const double p0  = 1.0 + (double)LAM_F;                // 11
    const double A   = (p0 * mu1 - P) / dsc;
    const double Bc  = p0 - A;
    const double l2r = log2((0.5 * (S - dsc)) / mu1);
    for (int i = tid; i < NTT; i += TPB) {
      double bi;
      if (i == 0) {
        bi = p0;
      } else {
        const int k = (i == NTT - 1) ? (NTT - 2) : i;
        const double rk  = exp2((double)k * l2r);
        const double rk1 = exp2((double)(k - 1) * l2r);
        const double bk  = mu1 * (A + Bc * rk) / (A + Bc * rk1);
        bi = (i == NTT - 1) ? (p0 - P / bk) : bk;
      }
      s_b[i] = (float)bi;
      if (i >= 1) {
        double bprev;
        if (i == 1) {
          bprev = p0;
        } else {
          const int k = i - 1;
          const double rk  = exp2((double)k * l2r);
          const double rk1 = exp2((double)(k - 1) * l2r);
          bprev = mu1 * (A + Bc * rk) / (A + Bc * rk1);
        }
        s_w[i] = (float)(-(double)LAM_F / bprev);
      } else {
        s_w[i] = 0.0f;
      }
    }
  }
  __syncthreads();

  const size_t trace_base = (size_t)b * NTT * NXX + (size_t)(x0 + tid);
  float* __restrict__ tvp = tv + trace_base;
  const float vmin2 = VMIN_F * VMIN_F, vmax2 = VMAX_F * VMAX_F;

  // ================= Pass A: Dix + forward elimination (d') ================
  float yprev = 0.0f, dprev = 0.0f;
  auto stepA = [&](int t, float vr) {
    const float vr2 = vr * vr;
    const float y   = vr2 * ((float)t * DT_F);
    float vi2 = (t == 0) ? vr2 : (y - yprev) / DT_F;
    vi2 = fminf(fmaxf(vi2, vmin2), vmax2);
    const float tvv = sqrtf(vi2 + EPS_F);
    const float d   = (t == 0) ? tvv : (tvv - s_w[t] * dprev);
    tvp[(size_t)t * NXX] = d;                 // d' staged in tv region (L2)
    yprev = y; dprev = d;
  };

#if USE_TDM
  const float* gin = rms + (size_t)b * NTT * NXX + x0;  // block tile base
  const bool issuer = (tid < 32);                        // wave 0 (wave32)
  if (issuer) {
    tdm_issue(gin, (unsigned)(uintptr_t)&s_in[0][0][0]); // prologue: chunk 0
  }
  for (int c = 0; c < NCHUNK; ++c) {
    if (issuer) {
      if (c + 1 < NCHUNK) {
        tdm_issue(gin + (size_t)(c + 1) * CH * NXX,
                  (unsigned)(uintptr_t)&s_in[(c + 1) & 1][0][0]);
        __builtin_amdgcn_s_wait_tensorcnt(1);   // oldest (chunk c) complete
      } else {
        __builtin_amdgcn_s_wait_tensorcnt(0);
      }
    }
    __syncthreads();                            // publish chunk c to block
    const int buf = c & 1;
    #pragma unroll 4
    for (int r = 0; r < CH; ++r) {
      stepA(c * CH + r, s_in[buf][r][tid]);
    }
    __syncthreads();                            // buffer reusable next iter
  }
#else
  for (int t = 0; t < NTT; ++t) {
    stepA(t, rms[trace_base + (size_t)t * NXX]);
  }
#endif

  // ============== Pass B: back substitution (time_vel_sm) ==================
  {
    float d = tvp[(size_t)(NTT - 1) * NXX];
    float xn = d / s_b[NTT - 1];
    tvp[(size_t)(NTT - 1) * NXX] = xn;
    for (int t = NTT - 2; t >= 0; --t) {
      if (t >= 16) __builtin_prefetch(&tvp[(size_t)(t - 16) * NXX], 0, 3);
      const float di = tvp[(size_t)t * NXX];
      const float xi = (di + LAM_F * xn) / s_b[t];
      tvp[(size_t)t * NXX] = xi;
      xn = xi;
    }
  }

  // ===== Pass C: cumsum depth curve + merge interpolation onto z-grid =====
  {
    float* __restrict__ dvp = dv + (size_t)b * NZZ * NXX + (size_t)(x0 + tid);
    float zc = 0.0f, zprev = 0.0f, vprev = 0.0f;
    float zl = 0.0f, vl = 0.0f;   // (z_curve[t-1], v[t-1]) at final t
    int zi = 0;
    for (int t = 0; t < NTT; ++t) {
      if (t + 16 < NTT) __builtin_prefetch(&tvp[(size_t)(t + 16) * NXX], 0, 3);
      const float v = tvp[(size_t)t * NXX];
      zl = zprev; vl = vprev;
      zc += 0.5f * v * DT_F;            // z_curve[t]
      if (t == 0) { zl = zc; vl = v; }  // idx0 clips to 0 -> w = 0
      while (zi < NZZ && ((float)zi * DZ_F) <= zc) {
        const float zg    = (float)zi * DZ_F;
        const float denom = fmaxf(zc - zl, EPS_F);
        float wgt = (zg - zl) / denom;
        wgt = fminf(fmaxf(wgt, 0.0f), 1.0f);
        dvp[(size_t)zi * NXX] = vl + wgt * (v - vl);
        ++zi;
      }
      zprev = zc; vprev = v;
    }
    // z beyond z_curve[-1]: idx1 = nt-1, idx0 = nt-2
    for (; zi < NZZ; ++zi) {
      const float zg    = (float)zi * DZ_F;
      const float denom = fmaxf(zprev - zl, EPS_F);
      float wgt = (zg - zl) / denom;
      wgt = fminf(fmaxf(wgt, 0.0f), 1.0f);
      dvp[(size_t)zi * NXX] = vl + wgt * (vprev - vl);
    }
  }
}

extern "C" void kernel_launch(void* const* d_in, const int* in_sizes, int n_in,
                              void* d_out, int out_size, void* d_ws, size_t ws_size,
                              hipStream_t stream) {
  (void)in_sizes; (void)n_in; (void)out_size; (void)d_ws; (void)ws_size;
  const float* rms = (const float*)d_in[0];
  float* dv = (float*)d_out;                               // (32,1,600,512)
  float* tv = dv + (size_t)NBB * NZZ * NXX;                // (32,1,2000,512)
  dim3 grid(NBB * NXX / TPB);                              // 64 blocks
  SmoothDix_78211354460885_kernel<<<grid, dim3(TPB), 0, stream>>>(rms, dv, tv);
}